// TransformerModel_14705968021921
// MI455X (gfx1250) — compile-verified
//
#include <hip/hip_runtime.h>
#include <hip/hip_bf16.h>
#include <math.h>

typedef __attribute__((ext_vector_type(2))) float v2f;
typedef __attribute__((ext_vector_type(4))) float v4f;
typedef __attribute__((ext_vector_type(8))) float v8f;
typedef __attribute__((ext_vector_type(4))) unsigned int u32x4;
typedef __attribute__((ext_vector_type(4))) int i32x4;
typedef __attribute__((ext_vector_type(8))) int i32x8;

#define SEQ      1024
#define DMODEL   512
#define NHEAD    8
#define DHEAD    64
#define NLAYER   12
#define DFF      2048
#define EPS_ATTN 1e-6f
#define EPS_LN   1e-5f

#ifndef USE_TDM
#define USE_TDM 1
#endif

enum { ACT_NONE = 0, ACT_PHI = 1, ACT_GELU = 2 };

// ---------------------------------------------------------------------------
// Tensor Data Mover: issue one 2D tile DMA (global -> LDS) per descriptor.
// D# layout per CDNA5 ISA (§8.3/§8.4):
//   group0: [1:0]=count=1, [63:32]=lds_addr(bytes), [120:64]=global_addr,
//           [127:126]=type=2
//   group1: [17:16]=data_size(2 -> 4B), [20]=pad_enable,
//           [24:22]=pad_interval (3 -> every 16 DW, 4 -> every 32 DW),
//           [31:25]=pad_amount (0 -> +1 DW)  => HW builds stride-17/33 LDS rows
//           [79:48]=tensor_dim0, [111:80]=tensor_dim1,
//           [127:112]=tile_dim0, [143:128]=tile_dim1, tile_dim2=0 (2D),
//           [207:160]=tensor_dim0_stride (elements)
// tensor_dims are set == tile dims (tiles never go OOB here).
// ---------------------------------------------------------------------------
__device__ __forceinline__ void tdm_load_2d(const float* gptr, unsigned lds_off,
                                            unsigned tile_d0, unsigned tile_d1,
                                            unsigned stride0_elems,
                                            unsigned pad_interval)
{
    unsigned long long ga = (unsigned long long)(size_t)gptr;
    u32x4 g0;
    g0[0] = 1u;                                     // count = 1, user mode
    g0[1] = lds_off;                                // LDS byte address
    g0[2] = (unsigned)ga;                           // global_addr[31:0]
    g0[3] = (unsigned)(ga >> 32) | (2u << 30);      // global_addr[56:32] | type=2
    i32x8 g1;
    g1[0] = (int)((2u << 16) | (1u << 20) | (pad_interval << 22)); // 4B, pad, +1DW
    g1[1] = (int)((tile_d0 & 0xffffu) << 16);       // tensor_dim0 lo16
    g1[2] = (int)((tile_d1 & 0xffffu) << 16);       // tensor_dim0 hi=0 | tensor_dim1 lo16
    g1[3] = (int)(tile_d0 << 16);                   // tensor_dim1 hi=0 | tile_dim0
    g1[4] = (int)tile_d1;                           // tile_dim1 | tile_dim2=0
    g1[5] = (int)stride0_elems;                     // dim0 stride lo32 (elements)
    g1[6] = 0;                                      // dim0 stride hi | dim1 stride lo
    g1[7] = 0;
    i32x4 gz = {0, 0, 0, 0};                        // groups 2/3: 2D tensor, unused
#if defined(__clang_major__) && (__clang_major__ >= 23)
    i32x8 gz8 = {0, 0, 0, 0, 0, 0, 0, 0};
    __builtin_amdgcn_tensor_load_to_lds(g0, g1, gz, gz, gz8, 0);
#else
    __builtin_amdgcn_tensor_load_to_lds(g0, g1, gz, gz, 0);
#endif
}

// ---------------------------------------------------------------------------
// Embedding gather + concat (6 tables -> 768 cols), scaled by sqrt(emb_size)
// ---------------------------------------------------------------------------
__global__ __launch_bounds__(256)
void embed_kernel(const int* __restrict__ x,
                  const float* __restrict__ e0, const float* __restrict__ e1,
                  const float* __restrict__ e2, const float* __restrict__ e3,
                  const float* __restrict__ e4, const float* __restrict__ e5,
                  float* __restrict__ E)
{
    const int s   = blockIdx.x;
    const int tid = threadIdx.x;
    const int* xs = x + s * 6;
    #pragma unroll
    for (int j = 0; j < 3; ++j) {
        int c = tid + j * 256;
        float v;
        if      (c < 32)  v = e0[xs[0]*32  +  c       ] * 5.656854249f;   // sqrt(32)
        else if (c < 160) v = e1[xs[1]*128 + (c - 32) ] * 11.313708499f;  // sqrt(128)
        else if (c < 416) v = e2[xs[2]*256 + (c - 160)] * 16.0f;          // sqrt(256)
        else if (c < 672) v = e3[xs[3]*256 + (c - 416)] * 16.0f;          // sqrt(256)
        else if (c < 704) v = e4[xs[4]*32  + (c - 672)] * 5.656854249f;   // sqrt(32)
        else              v = e5[xs[5]*64  + (c - 704)] * 8.0f;           // sqrt(64)
        E[s * 768 + c] = v;
    }
}

// ---------------------------------------------------------------------------
// Sinusoidal positional encoding, added in place
// ---------------------------------------------------------------------------
__global__ __launch_bounds__(256)
void add_pos_kernel(float* __restrict__ h)
{
    const int s   = blockIdx.x;
    const int tid = threadIdx.x;
    for (int c = tid; c < DMODEL; c += 256) {
        int   i2  = c & ~1;
        float div = __expf((float)i2 * (-9.210340372f / 512.0f)); // ln(10000)/d
        float ang = (float)s * div;
        float pe  = (c & 1) ? __cosf(ang) : __sinf(ang);
        h[s * DMODEL + c] += pe;
    }
}

// ---------------------------------------------------------------------------
// fp32 WMMA block GEMM:  C[M,N] = act(A[M,K] @ B[K,N] + bias) (+ residual)
// 256 threads = 8 waves; block tile 128x32; wave tile 16x32 (two 16x16 accs);
// K in chunks of 16 -> 4 x V_WMMA_F32_16X16X4_F32 per acc per chunk.
// Tiles are streamed into double-buffered LDS by the Tensor Data Mover
// (wave 0 issues, s_wait_tensorcnt + one barrier per chunk), so chunk k+1's
// DMA overlaps chunk k's WMMAs. HW pad gives stride-17/33 LDS rows.
// Requires M%128==0, N%32==0, K%16==0 (true for every shape here).
// ---------------------------------------------------------------------------
template<int ACT, bool RES>
__global__ __launch_bounds__(256)
void gemm_kernel(const float* __restrict__ A, const float* __restrict__ B,
                 const float* __restrict__ bias, const float* __restrict__ res,
                 float* __restrict__ C, int M, int N, int K)
{
    __shared__ float As[2][128 * 17];   // 128 rows x 16 K (+1 pad) x 2 buffers
    __shared__ float Bs[2][16 * 33];    // 16 K x 32 N (+1 pad) x 2 buffers

    const int tid  = threadIdx.x;
    const int lane = tid & 31;
    const int wid  = tid >> 5;
    const int hi   = lane >> 4;
    const int m16  = lane & 15;
    const int m0   = blockIdx.y * 128;
    const int n0   = blockIdx.x * 32;

    const float* Ablk = A + (size_t)m0 * K;
    const float* Bblk = B + n0;

    v8f acc0 = {};
    v8f acc1 = {};

#if USE_TDM
    if (tid < 32) {   // wave 0 drives the TDM
        tdm_load_2d(Ablk, (unsigned)(size_t)&As[0][0], 16u, 128u, (unsigned)K, 3u);
        tdm_load_2d(Bblk, (unsigned)(size_t)&Bs[0][0], 32u, 16u,  (unsigned)N, 4u);
    }
    int cur = 0;
#endif

    for (int k0 = 0; k0 < K; k0 += 16) {
#if USE_TDM
        if (tid < 32) __builtin_amdgcn_s_wait_tensorcnt(0);
        __syncthreads();                       // tiles visible to all waves
        if (tid < 32 && (k0 + 16) < K) {       // prefetch next chunk into other buffer
            tdm_load_2d(Ablk + (k0 + 16),
                        (unsigned)(size_t)&As[cur ^ 1][0], 16u, 128u, (unsigned)K, 3u);
            tdm_load_2d(Bblk + (size_t)(k0 + 16) * N,
                        (unsigned)(size_t)&Bs[cur ^ 1][0], 32u, 16u,  (unsigned)N, 4u);
        }
        const float* Asc = As[cur];
        const float* Bsc = Bs[cur];
#else
        // Fallback: vectorized staging (float4 global loads, many in flight)
        {
            const v4f* Ag = (const v4f*)(Ablk + k0);
            #pragma unroll
            for (int i = 0; i < 2; ++i) {
                int idx = tid + i * 256;                 // 512 float4 = 2048 floats
                int r = idx >> 2, c4 = idx & 3;
                v4f t = Ag[(size_t)r * (K >> 2) + c4];
                As[0][r * 17 + c4 * 4 + 0] = t.x;
                As[0][r * 17 + c4 * 4 + 1] = t.y;
                As[0][r * 17 + c4 * 4 + 2] = t.z;
                As[0][r * 17 + c4 * 4 + 3] = t.w;
            }
            if (tid < 128) {                             // 128 float4 = 512 floats
                int r = tid >> 3, c4 = tid & 7;
                const v4f* Bg = (const v4f*)(Bblk + (size_t)(k0 + r) * N);
                v4f t = Bg[c4];
                Bs[0][r * 33 + c4 * 4 + 0] = t.x;
                Bs[0][r * 33 + c4 * 4 + 1] = t.y;
                Bs[0][r * 33 + c4 * 4 + 2] = t.z;
                Bs[0][r * 33 + c4 * 4 + 3] = t.w;
            }
        }
        __syncthreads();
        const float* Asc = As[0];
        const float* Bsc = Bs[0];
#endif

        #pragma unroll
        for (int kk = 0; kk < 4; ++kk) {
            const int kl = kk * 4 + hi * 2;
            v2f a;                                   // A: (M=lane&15, K=2*half+j)
            a.x = Asc[(wid * 16 + m16) * 17 + kl];
            a.y = Asc[(wid * 16 + m16) * 17 + kl + 1];
            v2f b0, b1;                              // B: (K=2*half+j, N=lane&15)
            b0.x = Bsc[(kl)     * 33 + m16];
            b0.y = Bsc[(kl + 1) * 33 + m16];
            b1.x = Bsc[(kl)     * 33 + 16 + m16];
            b1.y = Bsc[(kl + 1) * 33 + 16 + m16];
            acc0 = __builtin_amdgcn_wmma_f32_16x16x4_f32(false, a, false, b0,
                                                         (short)0, acc0, false, false);
            acc1 = __builtin_amdgcn_wmma_f32_16x16x4_f32(false, a, false, b1,
                                                         (short)0, acc1, false, false);
        }

#if USE_TDM
        cur ^= 1;
#else
        __syncthreads();
#endif
    }

    // Epilogue: C/D layout VGPR r -> (M = r + 8*(lane>>4), N = lane&15)
    #pragma unroll
    for (int r = 0; r < 8; ++r) {
        const int m = m0 + wid * 16 + r + hi * 8;
        #pragma unroll
        for (int t = 0; t < 2; ++t) {
            const int n = n0 + t * 16 + m16;
            float v = (t ? acc1[r] : acc0[r]) + bias[n];
            if (ACT == ACT_PHI)  v = (v > 0.0f) ? (v + 1.0f) : expf(v);       // elu+1
            if (ACT == ACT_GELU) v = 0.5f * v * (1.0f + erff(v * 0.70710678118654752f));
            if (RES)             v += res[(size_t)m * N + n];
            C[(size_t)m * N + n] = v;
        }
    }
}

// ---------------------------------------------------------------------------
// Causal linear attention scan. One workgroup per head; 64x64 fp32 KV state
// resident in LDS (16.6 KB << 320 KB/WGP). Inclusive cumsum of k (x) v and k.
// ---------------------------------------------------------------------------
__global__ __launch_bounds__(256)
void attn_scan_kernel(const float* __restrict__ Q, const float* __restrict__ K,
                      const float* __restrict__ V, float* __restrict__ O)
{
    __shared__ float kv[64 * 65];
    __shared__ float ksum[64];
    __shared__ float qr[64], kr[64], vr[64];
    __shared__ float part[4 * 64];

    const int tid  = threadIdx.x;
    const int base = blockIdx.x * DHEAD;
    const int e    = tid & 63;
    const int g    = tid >> 6;

    for (int i = tid; i < 64 * 65; i += 256) kv[i] = 0.0f;
    if (tid < 64) ksum[tid] = 0.0f;
    __syncthreads();

    for (int t = 0; t < SEQ; ++t) {
        const int rowoff = t * DMODEL + base;
        if (tid < 64) {
            qr[tid] = Q[rowoff + tid];
            kr[tid] = K[rowoff + tid];
            vr[tid] = V[rowoff + tid];
        }
        __syncthreads();

        const float ve = vr[e];
        float acc = 0.0f;
        #pragma unroll
        for (int dd = 0; dd < 16; ++dd) {
            const int d = g * 16 + dd;
            float kvn = kv[d * 65 + e] + kr[d] * ve;
            kv[d * 65 + e] = kvn;
            acc += qr[d] * kvn;
        }
        part[g * 64 + e] = acc;
        if (tid < 64) ksum[tid] += kr[tid];
        __syncthreads();

        if (tid < 64) {
            float denom = EPS_ATTN;
            #pragma unroll 8
            for (int d = 0; d < 64; ++d) denom += qr[d] * ksum[d];
            float o = (part[tid] + part[64 + tid] + part[128 + tid] + part[192 + tid]) / denom;
            O[rowoff + tid] = o;
        }
        __syncthreads();
    }
}

// ---------------------------------------------------------------------------
// LayerNorm over 512 cols (block per row, LDS tree reduction)
// ---------------------------------------------------------------------------
__global__ __launch_bounds__(256)
void ln_kernel(const float* __restrict__ x, const float* __restrict__ g,
               const float* __restrict__ b, float* __restrict__ out)
{
    __shared__ float s1[256], s2[256];
    const int row = blockIdx.x, tid = threadIdx.x;
    const float* xr = x + (size_t)row * DMODEL;
    const float v0 = xr[tid], v1 = xr[tid + 256];
    s1[tid] = v0 + v1;
    s2[tid] = v0 * v0 + v1 * v1;
    __syncthreads();
    for (int off = 128; off > 0; off >>= 1) {
        if (tid < off) { s1[tid] += s1[tid + off]; s2[tid] += s2[tid + off]; }
        __syncthreads();
    }
    const float mu  = s1[0] * (1.0f / 512.0f);
    const float var = s2[0] * (1.0f / 512.0f) - mu * mu;
    const float inv = rsqrtf(var + EPS_LN);
    out[(size_t)row * DMODEL + tid]       = (v0 - mu) * inv * g[tid]       + b[tid];
    out[(size_t)row * DMODEL + tid + 256] = (v1 - mu) * inv * g[tid + 256] + b[tid + 256];
}

// ---------------------------------------------------------------------------
// Emotion projection [1024,512] @ [512,8] + b  (N=8 too narrow for WMMA tile)
// ---------------------------------------------------------------------------
__global__ __launch_bounds__(256)
void proj_kernel(const float* __restrict__ h, const float* __restrict__ w,
                 const float* __restrict__ b, float* __restrict__ out)
{
    const int idx = blockIdx.x * 256 + threadIdx.x;
    const int s = idx >> 3, j = idx & 7;
    float acc = b[j];
    for (int c = 0; c < DMODEL; ++c) acc += h[s * DMODEL + c] * w[c * 8 + j];
    out[idx] = acc;
}

// ---------------------------------------------------------------------------
extern "C" void kernel_launch(void* const* d_in, const int* in_sizes, int n_in,
                              void* d_out, int out_size, void* d_ws, size_t ws_size,
                              hipStream_t stream)
{
    (void)in_sizes; (void)n_in; (void)out_size; (void)ws_size;

    const int*   x    = (const int*)  d_in[0];
    const float* e0   = (const float*)d_in[1];
    const float* e1   = (const float*)d_in[2];
    const float* e2   = (const float*)d_in[3];
    const float* e3   = (const float*)d_in[4];
    const float* e4   = (const float*)d_in[5];
    const float* e5   = (const float*)d_in[6];
    const float* in_w = (const float*)d_in[7];
    const float* in_b = (const float*)d_in[8];
    const float* wq   = (const float*)d_in[9];
    const float* bq   = (const float*)d_in[10];
    const float* wk   = (const float*)d_in[11];
    const float* bk   = (const float*)d_in[12];
    const float* wv   = (const float*)d_in[13];
    const float* bv   = (const float*)d_in[14];
    const float* wo   = (const float*)d_in[15];
    const float* bo   = (const float*)d_in[16];
    const float* ln1g = (const float*)d_in[17];
    const float* ln1b = (const float*)d_in[18];
    const float* w1   = (const float*)d_in[19];
    const float* b1   = (const float*)d_in[20];
    const float* w2   = (const float*)d_in[21];
    const float* b2   = (const float*)d_in[22];
    const float* ln2g = (const float*)d_in[23];
    const float* ln2b = (const float*)d_in[24];
    const float* lnfg = (const float*)d_in[25];
    const float* lnfb = (const float*)d_in[26];
    const float* pw   = (const float*)d_in[27];
    const float* pb   = (const float*)d_in[28];

    // Workspace layout (floats). FFN buffer F (2,097,152) aliases the embedding
    // buffer E (786,432) -- E is dead after the input projection. Total 20 MB.
    float* ws = (float*)d_ws;
    float* F  = ws;
    float* E  = ws;
    float* H  = F  + (size_t)SEQ * DFF;
    float* Qb = H  + (size_t)SEQ * DMODEL;
    float* Kb = Qb + (size_t)SEQ * DMODEL;
    float* Vb = Kb + (size_t)SEQ * DMODEL;
    float* AO = Vb + (size_t)SEQ * DMODEL;
    float* T  = AO + (size_t)SEQ * DMODEL;

    const dim3 blk(256);
    const dim3 g512(DMODEL / 32, SEQ / 128);
    const dim3 gff (DFF   / 32, SEQ / 128);

    embed_kernel<<<SEQ, blk, 0, stream>>>(x, e0, e1, e2, e3, e4, e5, E);
    gemm_kernel<ACT_NONE, false><<<g512, blk, 0, stream>>>(E, in_w, in_b, nullptr, H,
                                                           SEQ, DMODEL, 768);
    add_pos_kernel<<<SEQ, blk, 0, stream>>>(H);

    for (int l = 0; l < NLAYER; ++l) {
        const float* wq_l = wq + (size_t)l * DMODEL * DMODEL;
        const float* wk_l = wk + (size_t)l * DMODEL * DMODEL;
        const float* wv_l = wv + (size_t)l * DMODEL * DMODEL;
        const float* wo_l = wo + (size_t)l * DMODEL * DMODEL;
        const float* w1_l = w1 + (size_t)l * DMODEL * DFF;
        const float* w2_l = w2 + (size_t)l * DFF * DMODEL;
        const float* bq_l = bq + (size_t)l * DMODEL;
        const float* bk_l = bk + (size_t)l * DMODEL;
        const float* bv_l = bv + (size_t)l * DMODEL;
        const float* bo_l = bo + (size_t)l * DMODEL;
        const float* b1_l = b1 + (size_t)l * DFF;
        const float* b2_l = b2 + (size_t)l * DMODEL;

        gemm_kernel<ACT_PHI,  false><<<g512, blk, 0, stream>>>(H, wq_l, bq_l, nullptr, Qb,
                                                               SEQ, DMODEL, DMODEL);
        gemm_kernel<ACT_PHI,  false><<<g512, blk, 0, stream>>>(H, wk_l, bk_l, nullptr, Kb,
                                                               SEQ, DMODEL, DMODEL);
        gemm_kernel<ACT_NONE, false><<<g512, blk, 0, stream>>>(H, wv_l, bv_l, nullptr, Vb,
                                                               SEQ, DMODEL, DMODEL);
        attn_scan_kernel<<<NHEAD, blk, 0, stream>>>(Qb, Kb, Vb, AO);
        gemm_kernel<ACT_NONE, true ><<<g512, blk, 0, stream>>>(AO, wo_l, bo_l, H, T,
                                                               SEQ, DMODEL, DMODEL);
        ln_kernel<<<SEQ, blk, 0, stream>>>(T, ln1g + l * DMODEL, ln1b + l * DMODEL, H);
        gemm_kernel<ACT_GELU, false><<<gff,  blk, 0, stream>>>(H, w1_l, b1_l, nullptr, F,
                                                               SEQ, DFF, DMODEL);
        gemm_kernel<ACT_NONE, true ><<<g512, blk, 0, stream>>>(F, w2_l, b2_l, H, T,
                                                               SEQ, DMODEL, DFF);
        ln_kernel<<<SEQ, blk, 0, stream>>>(T, ln2g + l * DMODEL, ln2b + l * DMODEL, H);
    }

    float* hout = (float*)d_out;
    ln_kernel<<<SEQ, blk, 0, stream>>>(H, lnfg, lnfb, hout);
    proj_kernel<<<32, blk, 0, stream>>>(hout, pw, pb, hout + (size_t)SEQ * DMODEL);
}